// LIFFeedForwardLayer_58892591563415
// MI455X (gfx1250) — compile-verified
//
#include <hip/hip_runtime.h>
#include <stdint.h>

// LIF scan: T=256 sequential steps over 32*4096 = 131072 neurons.
// Bandwidth-bound streaming scan -> CDNA5 async-to-LDS pipeline (ASYNCcnt),
// L2 prefetch (global_prefetch_b8), non-temporal spike stores.

typedef float v4f __attribute__((ext_vector_type(4)));

#define T_STEPS 256
#define BN      (32 * 4096)       // neurons per timestep
#define VEC     4                 // floats per thread
#define THREADS 256
#define NIDX    (BN / VEC)        // 32768 float4 slots per timestep
#define BLOCKS  (NIDX / THREADS)  // 128
#define NBUF    16                // async pipeline depth (timesteps in flight)

// LIF constants: dv = 0.1*((0 - v) + i) -> v_dec = 0.9v + 0.1i ; i_dec = 0.8i
#define A_V 0.9f
#define B_V 0.1f
#define A_I 0.8f

// Low 32 bits of a flat pointer to LDS == the LDS byte address (generic->local
// addrspacecast is a truncate on AMDGPU). The ptrtoint also escapes `smem`, so
// the memory-clobbering asms below are treated as potential writers of it and
// the ds_load reads cannot be folded away.
__device__ __forceinline__ uint32_t lds_addr_of(const void* p) {
  return (uint32_t)(uintptr_t)p;
}

__global__ __launch_bounds__(THREADS) void lif_async_kernel(
    const float* __restrict__ x, float* __restrict__ out) {
  // Ring of NBUF timestep slabs; each lane owns one 16B slot per slab.
  __shared__ v4f smem[NBUF * THREADS];  // 64 KB

  const int tid = threadIdx.x;
  const int idx = blockIdx.x * THREADS + tid;       // float4 index in [0, NIDX)
  const uint64_t step_bytes = (uint64_t)BN * 4u;    // bytes per timestep slab
  uint64_t gaddr = (uint64_t)(uintptr_t)x + (uint64_t)idx * 16u;

  // ---- prologue: fill the async pipeline with the first NBUF timesteps ----
#pragma unroll
  for (int p = 0; p < NBUF; ++p) {
    const uint32_t laddr = lds_addr_of(&smem[p * THREADS + tid]);
    asm volatile("global_load_async_to_lds_b128 %0, %1, off"
                 :: "v"(laddr), "v"(gaddr)
                 : "memory");
    gaddr += step_bytes;
  }

  v4f v   = {0.0f, 0.0f, 0.0f, 0.0f};
  v4f cur = {0.0f, 0.0f, 0.0f, 0.0f};
  v4f* out4 = (v4f*)out;

  int t = 0;
  // ---- steady state: consume oldest slab, refill it with step t+NBUF ----
  for (; t < T_STEPS - NBUF; ++t) {
    // Async loads retire in order: <=15 outstanding => oldest of 16 landed.
    asm volatile("s_wait_asynccnt 15" ::: "memory");
    const int buf = t & (NBUF - 1);
    v4f xv = smem[buf * THREADS + tid];               // ds_load_b128

    // Warm L2 ~24 steps ahead (gaddr is already NBUF steps ahead).
    __builtin_prefetch((const void*)(uintptr_t)(gaddr + 8u * step_bytes), 0, 0);

    v4f z;
#pragma unroll
    for (int k = 0; k < 4; ++k) {
      const float vd = A_V * v[k] + B_V * cur[k];
      const float id = A_I * cur[k];
      const bool sp  = vd > 1.0f;           // heaviside(vd - v_th, 0)
      z[k]   = sp ? 1.0f : 0.0f;
      v[k]   = sp ? 0.0f : vd;              // reset to 0 on spike
      cur[k] = id + xv[k];                  // inject input after decay
    }
    // Spikes are write-once: non-temporal, keep L2 for the x stream.
    __builtin_nontemporal_store(z, &out4[(size_t)t * NIDX + idx]);

    // xv consumed above -> safe to overwrite this lane's slot.
    const uint32_t laddr = lds_addr_of(&smem[buf * THREADS + tid]);
    asm volatile("global_load_async_to_lds_b128 %0, %1, off"
                 :: "v"(laddr), "v"(gaddr)
                 : "memory");
    gaddr += step_bytes;
  }

  // ---- tail: drain all outstanding async loads, consume remaining slabs ----
  asm volatile("s_wait_asynccnt 0" ::: "memory");
  for (; t < T_STEPS; ++t) {
    const int buf = t & (NBUF - 1);
    v4f xv = smem[buf * THREADS + tid];
    v4f z;
#pragma unroll
    for (int k = 0; k < 4; ++k) {
      const float vd = A_V * v[k] + B_V * cur[k];
      const float id = A_I * cur[k];
      const bool sp  = vd > 1.0f;
      z[k]   = sp ? 1.0f : 0.0f;
      v[k]   = sp ? 0.0f : vd;
      cur[k] = id + xv[k];
    }
    __builtin_nontemporal_store(z, &out4[(size_t)t * NIDX + idx]);
  }

  // ---- final state: v_T then i_T, appended after the T*BN spike block ----
  out4[(size_t)T_STEPS * NIDX + idx]        = v;
  out4[(size_t)T_STEPS * NIDX + NIDX + idx] = cur;
}

extern "C" void kernel_launch(void* const* d_in, const int* in_sizes, int n_in,
                              void* d_out, int out_size, void* d_ws, size_t ws_size,
                              hipStream_t stream) {
  (void)in_sizes; (void)n_in; (void)out_size; (void)d_ws; (void)ws_size;
  const float* x = (const float*)d_in[0];
  float* out = (float*)d_out;
  lif_async_kernel<<<BLOCKS, THREADS, 0, stream>>>(x, out);
}